// HeteroGNN_2413771620545
// MI455X (gfx1250) — compile-verified
//
#include <hip/hip_runtime.h>
#include <cmath>

// ---------------------------------------------------------------------------
// HeteroGNN forward for MI455X (gfx1250, wave32).
// G=128 graphs, NA=NV=256, Nt=32768, DA=128, DV=512, H=128, K=3, OUT=128.
// fp32 throughout; dense GEMMs via V_WMMA_F32_16X16X4_F32 with the shared
// weight panel staged into LDS by the Tensor Data Mover (TENSOR_LOAD_TO_LDS).
// ---------------------------------------------------------------------------

typedef float v2f __attribute__((ext_vector_type(2)));
typedef float v8f __attribute__((ext_vector_type(8)));
typedef unsigned int u32x4 __attribute__((ext_vector_type(4)));
typedef int i32x4 __attribute__((ext_vector_type(4)));
typedef int i32x8 __attribute__((ext_vector_type(8)));

#define NT_   32768
#define H_    128
#define G_    128
#define NPG_  256      // nodes per graph (both audio and video)

// ---------------- TDM: stage a [rows x 128] f32 panel into LDS -------------
// D# group0: count=1, lds_addr in [63:32], global_addr in [120:64], type=2.
// D# group1: data_size=4B, tensor_dim0=128, tensor_dim1=rows,
//            tile_dim0=128, tile_dim1=rows, tensor_dim0_stride=128.
// Groups 2/3 zero (2D tensor). Wave-level op, EXEC-independent.
// This toolchain declares the 6-arg builtin:
//   (u32x4 g0, i32x8 g1, i32x4 g2, i32x4 g3, i32x8 pad, i32 cpol)
__device__ __forceinline__ void tdm_load_panel(const float* gsrc,
                                               float* lds_dst, int rows) {
  unsigned long long ga = (unsigned long long)(const void*)gsrc;
  unsigned lds = (unsigned)(unsigned long long)(void*)lds_dst; // LDS offset
  u32x4 g0;
  g0.x = 1u;                                    // count=1, user mode
  g0.y = lds;                                   // lds_addr (bytes)
  g0.z = (unsigned)(ga & 0xffffffffull);        // global_addr[31:0]
  g0.w = (unsigned)((ga >> 32) & 0x01ffffffull) // global_addr[56:32]
         | (2u << 30);                          // type=2 ("image")
  i32x8 g1;
  g1[0] = (int)(2u << 16);                      // wg_mask=0, data_size=2 (4B)
  g1[1] = (int)(128u << 16);                    // tensor_dim0[15:0]=128
  g1[2] = (int)(((unsigned)rows & 0xffffu) << 16); // tensor_dim1[15:0]=rows
  g1[3] = (int)(128u << 16);                    // tile_dim0=128
  g1[4] = (int)(unsigned)rows;                  // tile_dim1=rows, tile_dim2=0
  g1[5] = 128;                                  // tensor_dim0_stride=128
  g1[6] = 0;
  g1[7] = 0;
  i32x4 z4 = {0, 0, 0, 0};
  i32x8 z8 = {0, 0, 0, 0, 0, 0, 0, 0};
  __builtin_amdgcn_tensor_load_to_lds(g0, g1, z4, z4, z8, 0);
}

// ---------------- fp32 WMMA GEMM: C[M,128] = A[M,K] @ B[K,128] -------------
// One block = 8 waves; wave w computes rows [blk*128 + w*16, +16) x all 128
// cols. B is staged chunk-wise (128 rows) into LDS by the TDM; A streams
// through the global path. Grid must tile M exactly (M % 128 == 0).
__global__ __launch_bounds__(256) void wmma_gemm_f32(
    const float* __restrict__ A, const float* __restrict__ B,
    float* __restrict__ C, int M, int K) {
  __shared__ float bsh[128 * H_];   // 64 KB B chunk
  const int lane = threadIdx.x & 31;
  const int wave = threadIdx.x >> 5;
  const int m0   = (blockIdx.x * 8 + wave) * 16;
  const int hf = lane >> 4;   // lane half: 0 -> lanes 0-15, 1 -> lanes 16-31
  const int lm = lane & 15;

  v8f acc[8];
  const v8f vzero = {0.f, 0.f, 0.f, 0.f, 0.f, 0.f, 0.f, 0.f};
#pragma unroll
  for (int nt = 0; nt < 8; ++nt) acc[nt] = vzero;

  const float* arow = A + (size_t)(m0 + lm) * K;   // A row for this lane's M

  for (int kc = 0; kc < K; kc += 128) {
    if (wave == 0) {
      tdm_load_panel(B + (size_t)kc * H_, bsh, 128);
      __builtin_amdgcn_s_wait_tensorcnt(0);
    }
    __syncthreads();
#pragma unroll 4
    for (int k = 0; k < 128; k += 4) {
      // 32-bit A 16x4 layout: VGPR j, half h holds K = k + j + 2*h
      v2f a;
      a.x = arow[kc + k + 2 * hf + 0];
      a.y = arow[kc + k + 2 * hf + 1];
#pragma unroll
      for (int nt = 0; nt < 8; ++nt) {
        // 32-bit B 4x16 layout (mirrored): VGPR j, half h -> row K=k+j+2h, N=lm
        v2f b;
        b.x = bsh[(k + 2 * hf + 0) * H_ + nt * 16 + lm];
        b.y = bsh[(k + 2 * hf + 1) * H_ + nt * 16 + lm];
        acc[nt] = __builtin_amdgcn_wmma_f32_16x16x4_f32(
            false, a, false, b, (short)0, acc[nt], false, false);
      }
    }
    __syncthreads();   // protect LDS before next chunk overwrites it
  }
  // C/D 16x16 f32 layout: VGPR j, lanes 16-31 -> M = j + 8, N = lm
#pragma unroll
  for (int nt = 0; nt < 8; ++nt)
#pragma unroll
    for (int j = 0; j < 8; ++j)
      C[(size_t)(m0 + j + 8 * hf) * H_ + nt * 16 + lm] = acc[nt][j];
}

// ---------------- utility: fill --------------------------------------------
__global__ void fill_f32(float* __restrict__ p, float v, unsigned n) {
  unsigned i = blockIdx.x * blockDim.x + threadIdx.x;
  if (i < n) p[i] = v;
}

// ---------------- edge scatter-add (GCN aggregate) -------------------------
// 32 lanes per edge; each lane handles 4 features via float4 gather + atomics.
__device__ __forceinline__ void atomAddF(float* p, float v) {
  __hip_atomic_fetch_add(p, v, __ATOMIC_RELAXED, __HIP_MEMORY_SCOPE_AGENT);
}

__global__ __launch_bounds__(256) void gcn_scatter(
    const float* __restrict__ h, const int* __restrict__ src,
    const int* __restrict__ dst, float* __restrict__ acc, int E) {
  int idx = blockIdx.x * 256 + threadIdx.x;
  int e = idx >> 5;
  if (e >= E) return;
  int lane = idx & 31;
  int s = src[e], d = dst[e];
  const float4 v = *(const float4*)(h + (size_t)s * H_ + lane * 4);
  float* p = acc + (size_t)d * H_ + lane * 4;
  atomAddF(p + 0, v.x); atomAddF(p + 1, v.y);
  atomAddF(p + 2, v.z); atomAddF(p + 3, v.w);
}

// ---------------- relu + graph LayerNorm (per-graph stats) -----------------
__global__ __launch_bounds__(256) void relu_graph_ln(
    const float* __restrict__ x, const float* __restrict__ bias1,
    const float* __restrict__ bias2, const float* __restrict__ gamma,
    const float* __restrict__ beta, float* __restrict__ out) {
  const int g = blockIdx.x, tid = threadIdx.x;
  const int T = NPG_ * H_;
  const float* xg = x + (size_t)g * T;
  float sum = 0.f, ss = 0.f;
  for (int i = tid; i < T; i += 256) {
    float b = bias1[i & (H_ - 1)] + (bias2 ? bias2[i & (H_ - 1)] : 0.f);
    float v = xg[i] + b; v = v > 0.f ? v : 0.f;
    sum += v; ss += v * v;
  }
  __shared__ float r1[256], r2[256];
  r1[tid] = sum; r2[tid] = ss; __syncthreads();
  for (int s = 128; s > 0; s >>= 1) {
    if (tid < s) { r1[tid] += r1[tid + s]; r2[tid] += r2[tid + s]; }
    __syncthreads();
  }
  const float mean = r1[0] / T;
  const float var  = r2[0] / T - mean * mean;
  const float rstd = rsqrtf(var + 1e-5f);
  float* og = out + (size_t)g * T;
  for (int i = tid; i < T; i += 256) {
    int f = i & (H_ - 1);
    float b = bias1[f] + (bias2 ? bias2[f] : 0.f);
    float v = xg[i] + b; v = v > 0.f ? v : 0.f;
    og[i] = (v - mean) * rstd * gamma[f] + beta[f];
  }
}

// ---------------- per-graph kNN (video -> 3 nearest audio) -----------------
// Rank by |a|^2 - 2 v.a (|v|^2 is row-constant). One thread per video node;
// video row register-cached; audio row loads are wave-uniform broadcasts.
__global__ __launch_bounds__(256) void knn_topk(
    const float* __restrict__ xa, const float* __restrict__ xv,
    int* __restrict__ knn_dst) {
  const int g = blockIdx.x, v = threadIdx.x;
  const int base = g * NPG_;
  __shared__ float na2[NPG_];
  {
    const float* ar = xa + (size_t)(base + v) * H_;
    float s = 0.f;
#pragma unroll
    for (int d = 0; d < H_; ++d) { float t = ar[d]; s += t * t; }
    na2[v] = s;
  }
  __syncthreads();
  float xvr[H_];
  const float* vr = xv + (size_t)(base + v) * H_;
#pragma unroll
  for (int d = 0; d < H_; ++d) xvr[d] = vr[d];

  float t0 = 1e30f, t1 = 1e30f, t2 = 1e30f;
  int   i0 = 0, i1 = 0, i2 = 0;
  for (int a = 0; a < NPG_; ++a) {
    const float* ar = xa + (size_t)(base + a) * H_;
    float dot = 0.f;
#pragma unroll
    for (int d = 0; d < H_; ++d) dot += xvr[d] * ar[d];
    float sc = na2[a] - 2.f * dot;
    if (sc < t0)      { t2 = t1; i2 = i1; t1 = t0; i1 = i0; t0 = sc; i0 = a; }
    else if (sc < t1) { t2 = t1; i2 = i1; t1 = sc; i1 = a; }
    else if (sc < t2) { t2 = sc; i2 = a; }
  }
  int* o = knn_dst + (size_t)(base + v) * 3;
  o[0] = base + i0; o[1] = base + i1; o[2] = base + i2;
}

// ---------------- GEMV: out[n] = h[n,:] . a -------------------------------
__global__ __launch_bounds__(256) void gemv_dot(
    const float* __restrict__ h, const float* __restrict__ a,
    float* __restrict__ out, int N) {
  int n = blockIdx.x * 256 + threadIdx.x;
  if (n >= N) return;
  const float* r = h + (size_t)n * H_;
  float acc = 0.f;
#pragma unroll
  for (int d = 0; d < H_; ++d) acc += r[d] * a[d];
  out[n] = acc;
}

// ---------------- GAT over kNN edges ---------------------------------------
__device__ __forceinline__ void atomicMaxF(float* addr, float val) {
  int old = __float_as_int(*addr);
  while (__int_as_float(old) < val) {
    int prev = atomicCAS((int*)addr, old, __float_as_int(val));
    if (prev == old) break;
    old = prev;
  }
}

__global__ void gat_logits(const float* __restrict__ gs,
                           const float* __restrict__ gd,
                           const int* __restrict__ knn_dst,
                           float* __restrict__ ebuf, float* __restrict__ m,
                           int E) {
  int e = blockIdx.x * 256 + threadIdx.x;
  if (e >= E) return;
  int s = e / 3;           // src video node id (edge e = src*3 + k)
  int d = knn_dst[e];
  float x = gs[s] + gd[d];
  x = x > 0.f ? x : 0.2f * x;   // leaky_relu(0.2)
  ebuf[e] = x;
  atomicMaxF(&m[d], x);
}

__global__ void gat_softmax(float* __restrict__ ebuf,
                            const float* __restrict__ m,
                            const int* __restrict__ knn_dst,
                            float* __restrict__ ssum, int E) {
  int e = blockIdx.x * 256 + threadIdx.x;
  if (e >= E) return;
  int d = knn_dst[e];
  float ex = __expf(ebuf[e] - m[d]);
  ebuf[e] = ex;
  atomAddF(&ssum[d], ex);
}

__global__ __launch_bounds__(256) void gat_scatter(
    const float* __restrict__ ebuf, const float* __restrict__ ssum,
    const int* __restrict__ knn_dst, const float* __restrict__ hs,
    float* __restrict__ acc, int E) {
  int idx = blockIdx.x * 256 + threadIdx.x;
  int e = idx >> 5;
  if (e >= E) return;
  int lane = idx & 31;
  int s = e / 3;
  int d = knn_dst[e];
  float alpha = ebuf[e] / (ssum[d] + 1e-16f);
  const float4 v = *(const float4*)(hs + (size_t)s * H_ + lane * 4);
  float* p = acc + (size_t)d * H_ + lane * 4;
  atomAddF(p + 0, alpha * v.x); atomAddF(p + 1, alpha * v.y);
  atomAddF(p + 2, alpha * v.z); atomAddF(p + 3, alpha * v.w);
}

// ---------------- GlobalAttention readout + final linear -------------------
__global__ __launch_bounds__(256) void readout(
    const float* __restrict__ xa2, const float* __restrict__ att_w,
    const float* __restrict__ linW, const float* __restrict__ linb,
    float* __restrict__ out) {
  const int g = blockIdx.x, tid = threadIdx.x;
  __shared__ float r1[256];
  __shared__ float wsh[256];
  __shared__ float emb[H_];
  const float* row = xa2 + ((size_t)g * NPG_ + tid) * H_;
  float gate = 0.f;
#pragma unroll
  for (int d = 0; d < H_; ++d) gate += row[d] * att_w[d];
  r1[tid] = gate; __syncthreads();
  for (int s = 128; s > 0; s >>= 1) {
    if (tid < s) r1[tid] = fmaxf(r1[tid], r1[tid + s]);
    __syncthreads();
  }
  const float mx = r1[0]; __syncthreads();
  const float ex = __expf(gate - mx);
  wsh[tid] = ex; r1[tid] = ex; __syncthreads();
  for (int s = 128; s > 0; s >>= 1) {
    if (tid < s) r1[tid] += r1[tid + s];
    __syncthreads();
  }
  const float inv = 1.f / r1[0];
  if (tid < H_) {
    const float* xg = xa2 + (size_t)g * NPG_ * H_;
    float acc = 0.f;
    for (int n = 0; n < NPG_; ++n) acc += wsh[n] * xg[(size_t)n * H_ + tid];
    emb[tid] = acc * inv;
  }
  __syncthreads();
  if (tid < H_) {
    float acc = linb[tid];
#pragma unroll 8
    for (int d = 0; d < H_; ++d) acc += emb[d] * linW[d * H_ + tid];
    out[g * H_ + tid] = acc;
  }
}

// ---------------------------------------------------------------------------
extern "C" void kernel_launch(void* const* d_in, const int* in_sizes, int n_in,
                              void* d_out, int out_size, void* d_ws, size_t ws_size,
                              hipStream_t stream) {
  const float* x_audio = (const float*)d_in[0];
  const float* x_video = (const float*)d_in[1];
  const int*   edge_aa = (const int*)d_in[2];
  const int*   edge_vv = (const int*)d_in[3];
  const float* W0_a = (const float*)d_in[4];
  const float* b0_a = (const float*)d_in[5];
  const float* W0_v = (const float*)d_in[6];
  const float* b0_v = (const float*)d_in[7];
  const float* W1   = (const float*)d_in[8];
  const float* b1   = (const float*)d_in[9];
  const float* Ws   = (const float*)d_in[10];
  const float* Wd   = (const float*)d_in[11];
  const float* a_s  = (const float*)d_in[12];
  const float* a_d  = (const float*)d_in[13];
  const float* b_gat = (const float*)d_in[14];
  const float* ln    = (const float*)d_in[15];  // [2,2,2,128]
  const float* att_w = (const float*)d_in[16];
  const float* lin_W = (const float*)d_in[17];
  const float* lin_b = (const float*)d_in[18];

  const int NT = NT_, DA = 128, DV = 512;
  const int E_aa = in_sizes[2] / 2, E_vv = in_sizes[3] / 2;
  const int E_knn = NT * 3;
  const size_t FB = (size_t)NT * H_;   // 4M floats per feature buffer

  // workspace layout (floats): 5 big 16MB buffers + small scratch (~1.3MB)
  float* B1 = (float*)d_ws;            // h_a0 -> xa
  float* B2 = B1 + FB;                 // h_v0 -> xv -> acc_layer1(audio)
  float* B3 = B2 + FB;                 // acc_a0 -> h_w1a -> xa2
  float* B4 = B3 + FB;                 // acc_v0 -> hs
  float* B5 = B4 + FB;                 // hd
  float* gs   = B5 + FB;
  float* gd   = gs + NT;
  float* mbuf = gd + NT;
  float* sbuf = mbuf + NT;
  float* ebuf = sbuf + NT;             // E_knn floats
  int*   knn  = (int*)(ebuf + E_knn);  // E_knn ints

  const dim3 blk(256);
  const unsigned gGemm   = NT / 128;                    // 256 blocks
  const unsigned gFill   = (unsigned)((FB + 255) / 256);
  const unsigned gScatAA = (unsigned)((size_t)E_aa * 32 / 256);
  const unsigned gScatVV = (unsigned)((size_t)E_vv * 32 / 256);

  // ---- layer 0: x@W via WMMA (TDM-staged B), edge scatter, relu+graph-LN ----
  wmma_gemm_f32<<<gGemm, blk, 0, stream>>>(x_audio, W0_a, B1, NT, DA);
  wmma_gemm_f32<<<gGemm, blk, 0, stream>>>(x_video, W0_v, B2, NT, DV);
  fill_f32<<<gFill, blk, 0, stream>>>(B3, 0.f, (unsigned)FB);
  fill_f32<<<gFill, blk, 0, stream>>>(B4, 0.f, (unsigned)FB);
  gcn_scatter<<<gScatAA, blk, 0, stream>>>(B1, edge_aa, edge_aa + E_aa, B3, E_aa);
  gcn_scatter<<<gScatVV, blk, 0, stream>>>(B2, edge_vv, edge_vv + E_vv, B4, E_vv);
  relu_graph_ln<<<G_, blk, 0, stream>>>(B3, b0_a, nullptr, ln + 0,   ln + 128, B1);
  relu_graph_ln<<<G_, blk, 0, stream>>>(B4, b0_v, nullptr, ln + 256, ln + 384, B2);

  // ---- kNN matching graph (video -> audio) ----
  knn_topk<<<G_, blk, 0, stream>>>(B1, B2, knn);

  // ---- layer 1 projections (video GCN branch is dead in the reference) ----
  wmma_gemm_f32<<<gGemm, blk, 0, stream>>>(B1, W1, B3, NT, H_);  // h_w1a
  wmma_gemm_f32<<<gGemm, blk, 0, stream>>>(B2, Ws, B4, NT, H_);  // hs
  wmma_gemm_f32<<<gGemm, blk, 0, stream>>>(B1, Wd, B5, NT, H_);  // hd
  gemv_dot<<<NT / 256, blk, 0, stream>>>(B4, a_s, gs, NT);
  gemv_dot<<<NT / 256, blk, 0, stream>>>(B5, a_d, gd, NT);

  // ---- layer 1 aggregation: GCN scatter + GAT into the same accumulator ----
  fill_f32<<<gFill, blk, 0, stream>>>(B2, 0.f, (unsigned)FB);     // xv now dead
  fill_f32<<<(NT + 255) / 256, blk, 0, stream>>>(mbuf, -INFINITY, NT);
  fill_f32<<<(NT + 255) / 256, blk, 0, stream>>>(sbuf, 0.f, NT);
  gcn_scatter<<<gScatAA, blk, 0, stream>>>(B3, edge_aa, edge_aa + E_aa, B2, E_aa);
  gat_logits<<<(E_knn + 255) / 256, blk, 0, stream>>>(gs, gd, knn, ebuf, mbuf, E_knn);
  gat_softmax<<<(E_knn + 255) / 256, blk, 0, stream>>>(ebuf, mbuf, knn, sbuf, E_knn);
  gat_scatter<<<(unsigned)((size_t)E_knn * 32 / 256), blk, 0, stream>>>(
      ebuf, sbuf, knn, B4, B2, E_knn);
  relu_graph_ln<<<G_, blk, 0, stream>>>(B2, b1, b_gat, ln + 512, ln + 640, B3);

  // ---- global-attention readout + final linear ----
  readout<<<G_, blk, 0, stream>>>(B3, att_w, lin_W, lin_b, (float*)d_out);
}